// GNNModelDGL_10015863734373
// MI455X (gfx1250) — compile-verified
//
#include <hip/hip_runtime.h>
#include <hip/hip_bf16.h>
#include <math.h>

typedef float v2f __attribute__((ext_vector_type(2)));
typedef float v8f __attribute__((ext_vector_type(8)));

// ---------------- helpers ----------------

__device__ __forceinline__ void fadd_agent(float* p, float v) {
    __hip_atomic_fetch_add(p, v, __ATOMIC_RELAXED, __HIP_MEMORY_SCOPE_AGENT);
}

// order-preserving encode of float into uint for atomic umax segment-max
__device__ __forceinline__ unsigned encodeF(float x) {
    unsigned u = __float_as_uint(x);
    return (u & 0x80000000u) ? ~u : (u | 0x80000000u);
}
__device__ __forceinline__ float decodeF(unsigned u) {
    return (u & 0x80000000u) ? __uint_as_float(u & 0x7fffffffu)
                             : __uint_as_float(~u);
}

__device__ __forceinline__ float lrelu02(float v) { return v > 0.0f ? v : 0.2f * v; }

// ---------------- fp32 WMMA GEMM: C[nRows,128] = X[nRows,128] @ W[128,128] ----------------
// one wave per 16x16 output tile; nRows must be a multiple of 16.
__global__ __launch_bounds__(32)
void gemm128_wmma_f32(const float* __restrict__ X, const float* __restrict__ W,
                      float* __restrict__ C) {
    const int tileN = blockIdx.x;           // 0..7
    const int tileM = blockIdx.y;
    const int lane  = threadIdx.x;          // 0..31
    const int half  = lane >> 4;            // 0: K pair {0,1}, 1: K pair {2,3}
    const int l15   = lane & 15;
    const int m0 = tileM * 16, n0 = tileN * 16;
    const float* xrow = X + (size_t)(m0 + l15) * 128;
    const int col = n0 + l15;

    v8f acc = {};
    #pragma unroll 4
    for (int kk = 0; kk < 128; kk += 4) {
        const int ka = kk + half * 2;
        v2f a, b;
        a.x = xrow[ka];
        a.y = xrow[ka + 1];
        b.x = W[(size_t)ka * 128 + col];
        b.y = W[(size_t)(ka + 1) * 128 + col];
        acc = __builtin_amdgcn_wmma_f32_16x16x4_f32(false, a, false, b,
                                                    (short)0, acc, false, false);
    }
    const int rbase = m0 + half * 8;
    #pragma unroll
    for (int v = 0; v < 8; ++v)
        C[(size_t)(rbase + v) * 128 + col] = acc[v];
}

// ---------------- per-node attention coefficients el/er ----------------
__global__ void attn_coef(const float* __restrict__ feat,
                          const float* __restrict__ al, const float* __restrict__ ar,
                          float* __restrict__ el, float* __restrict__ er,
                          int n, int heads, int d) {
    int tid = blockIdx.x * blockDim.x + threadIdx.x;
    if (tid >= n * heads) return;
    int node = tid / heads, h = tid - node * heads;
    const float* f = feat + (size_t)node * 128 + h * d;
    const float* a = al + h * d;
    const float* r = ar + h * d;
    float sl = 0.0f, sr = 0.0f;
    for (int i = 0; i < d; ++i) { float v = f[i]; sl += v * a[i]; sr += v * r[i]; }
    el[tid] = sl;
    er[tid] = sr;
}

// ---------------- edge softmax: max pass ----------------
__global__ void edge_max(const int* __restrict__ src, const int* __restrict__ dst,
                         const float* __restrict__ el, const float* __restrict__ er,
                         unsigned* __restrict__ em, int nE, int heads) {
    int tid = blockIdx.x * blockDim.x + threadIdx.x;
    if (tid >= nE * heads) return;
    int e = tid / heads, h = tid - e * heads;
    int s = src[e], d = dst[e];
    float v = lrelu02(el[s * heads + h] + er[d * heads + h]);
    atomicMax(&em[d * heads + h], encodeF(v));
}

// ---------------- edge softmax: denominator pass ----------------
__global__ void edge_den(const int* __restrict__ src, const int* __restrict__ dst,
                         const float* __restrict__ el, const float* __restrict__ er,
                         const unsigned* __restrict__ em, float* __restrict__ den,
                         int nE, int heads) {
    int tid = blockIdx.x * blockDim.x + threadIdx.x;
    if (tid >= nE * heads) return;
    int e = tid / heads, h = tid - e * heads;
    int s = src[e], d = dst[e];
    float v = lrelu02(el[s * heads + h] + er[d * heads + h]);
    float ex = expf(v - decodeF(em[d * heads + h]));
    fadd_agent(&den[d * heads + h], ex);
}

// ---------------- weighted aggregation: agg[dst,f] += feat[src,f]*alpha ----------------
template <int DSHIFT>   // head = f >> DSHIFT  (DSHIFT=4 -> D=16, DSHIFT=7 -> D=128)
__global__ void edge_agg(const int* __restrict__ src, const int* __restrict__ dst,
                         const float* __restrict__ el, const float* __restrict__ er,
                         const unsigned* __restrict__ em, const float* __restrict__ den,
                         const float* __restrict__ feat, float* __restrict__ agg,
                         int nE, int heads) {
    int tid = blockIdx.x * blockDim.x + threadIdx.x;
    int e = tid >> 7;
    if (e >= nE) return;
    int f = tid & 127;
    int h = f >> DSHIFT;
    int s = src[e], d = dst[e];
    float v = lrelu02(el[s * heads + h] + er[d * heads + h]);
    float alpha = expf(v - decodeF(em[d * heads + h])) / den[d * heads + h];
    fadd_agent(&agg[(size_t)d * 128 + f], feat[(size_t)s * 128 + f] * alpha);
}

// ---------------- layer-1 finalize: h = ELU(agg + b1) ----------------
__global__ void finalize_elu(const float* __restrict__ agg, const float* __restrict__ b,
                             float* __restrict__ out, int n) {
    int tid = blockIdx.x * blockDim.x + threadIdx.x;
    if (tid >= n * 128) return;
    float x = agg[tid] + b[tid & 127];
    out[tid] = x > 0.0f ? x : expm1f(x);
}

// ---------------- layer-2 finalize: logits = agg + h (residual) + b2 ----------------
__global__ void finalize_res(const float* __restrict__ agg, const float* __restrict__ hres,
                             const float* __restrict__ b, float* __restrict__ out, int n) {
    int tid = blockIdx.x * blockDim.x + threadIdx.x;
    if (tid >= n * 128) return;
    out[tid] = agg[tid] + hres[tid] + b[tid & 127];
}

// ---------------- per-graph mean pooling (accumulate) ----------------
__global__ void pool_acc(const float* __restrict__ logits, const int* __restrict__ gid,
                         float* __restrict__ gsum, float* __restrict__ gcnt, int n) {
    int tid = blockIdx.x * blockDim.x + threadIdx.x;
    int node = tid >> 7;
    if (node >= n) return;
    int f = tid & 127;
    int g = gid[node];
    fadd_agent(&gsum[(size_t)g * 128 + f], logits[(size_t)node * 128 + f]);
    if (f == 0) fadd_agent(&gcnt[g], 1.0f);
}

// ---------------- embedding mean + L2-normalize -> xn ----------------
__global__ void emb_norm(const float* __restrict__ gsum, const float* __restrict__ gcnt,
                         float* __restrict__ xn, int g) {
    int i = blockIdx.x * blockDim.x + threadIdx.x;
    if (i >= g) return;
    float cnt = fmaxf(gcnt[i], 1.0f);
    const float* row = gsum + (size_t)i * 128;
    float ss = 0.0f;
    for (int f = 0; f < 128; ++f) { float m = row[f] / cnt; ss += m * m; }
    float inv = 1.0f / sqrtf(ss);
    float* o = xn + (size_t)i * 128;
    for (int f = 0; f < 128; ++f) o[f] = (row[f] / cnt) * inv;
}

// ---------------- normalized-transposed ArcFace weight: wnT[k,o] = Wa[o,k]/||Wa_o|| ----------------
__global__ void wa_norm_t(const float* __restrict__ Wa, float* __restrict__ wnT) {
    int o = blockIdx.x * blockDim.x + threadIdx.x;
    if (o >= 128) return;
    const float* row = Wa + (size_t)o * 128;
    float ss = 0.0f;
    for (int k = 0; k < 128; ++k) ss += row[k] * row[k];
    float inv = 1.0f / sqrtf(ss);
    for (int k = 0; k < 128; ++k) wnT[(size_t)k * 128 + o] = row[k] * inv;
}

// ---------------- ArcFace margin + scale ----------------
__global__ void arcface(const float* __restrict__ cos1, const int* __restrict__ labels,
                        float* __restrict__ out, int g) {
    int tid = blockIdx.x * blockDim.x + threadIdx.x;
    int i = tid >> 7;
    if (i >= g) return;
    int o = tid & 127;
    float c = cos1[tid];
    c = fminf(1.0f, fmaxf(-1.0f, c));
    float m = (labels[i] == o) ? 0.5f : 0.0f;
    out[tid] = cosf(acosf(c) + m) * 4.0f;
}

// ---------------- host driver ----------------
extern "C" void kernel_launch(void* const* d_in, const int* in_sizes, int n_in,
                              void* d_out, int out_size, void* d_ws, size_t ws_size,
                              hipStream_t stream) {
    const float* x   = (const float*)d_in[0];
    const int*   src = (const int*)d_in[1];
    const int*   dst = (const int*)d_in[2];
    const int*   gid = (const int*)d_in[3];
    const int*   lab = (const int*)d_in[4];
    const float* W1  = (const float*)d_in[5];
    const float* al1 = (const float*)d_in[6];
    const float* ar1 = (const float*)d_in[7];
    const float* b1  = (const float*)d_in[8];
    const float* W2  = (const float*)d_in[9];
    const float* al2 = (const float*)d_in[10];
    const float* ar2 = (const float*)d_in[11];
    const float* b2  = (const float*)d_in[12];
    const float* Wa  = (const float*)d_in[13];
    float* out = (float*)d_out;

    const int N = in_sizes[0] / 128;
    const int E = in_sizes[1];
    const int G = in_sizes[4];
    const int H1 = 8;

    // workspace layout (floats)
    float* ws = (float*)d_ws;
    float*    FEAT = ws;                       // N*128  (feat1, feat2, then logits)
    float*    HBUF = FEAT + (size_t)N * 128;   // N*128  (h after layer 1)
    float*    AGG  = HBUF + (size_t)N * 128;   // N*128  (aggregation accumulator)
    float*    EL   = AGG  + (size_t)N * 128;   // N*8
    float*    ER   = EL   + (size_t)N * H1;    // N*8
    unsigned* EM   = (unsigned*)(ER + (size_t)N * H1); // N*8
    float*    DN   = (float*)(EM + (size_t)N * H1);    // N*8
    float*    GS   = DN + (size_t)N * H1;      // G*128
    float*    GC   = GS + (size_t)G * 128;     // G
    float*    XN   = GC + G;                   // G*128
    float*    WNT  = XN + (size_t)G * 128;     // 128*128
    float*    COS1 = WNT + 128 * 128;          // G*128

    const int TB = 256;
    const int nfBlocks  = (N * 128 + TB - 1) / TB;          // N*128 threads
    const int ehBlocks1 = (E * H1 + TB - 1) / TB;           // E*H threads, layer 1
    const int ehBlocks2 = (E + TB - 1) / TB;                // E threads, layer 2
    const int eaBlocks  = (int)(((long long)E * 128 + TB - 1) / TB);

    // ===== layer 1 =====
    gemm128_wmma_f32<<<dim3(8, N / 16), 32, 0, stream>>>(x, W1, FEAT);
    attn_coef<<<(N * H1 + TB - 1) / TB, TB, 0, stream>>>(FEAT, al1, ar1, EL, ER, N, H1, 16);
    hipMemsetAsync(EM, 0, (size_t)N * H1 * sizeof(unsigned), stream);
    hipMemsetAsync(DN, 0, (size_t)N * H1 * sizeof(float), stream);
    edge_max<<<ehBlocks1, TB, 0, stream>>>(src, dst, EL, ER, EM, E, H1);
    edge_den<<<ehBlocks1, TB, 0, stream>>>(src, dst, EL, ER, EM, DN, E, H1);
    hipMemsetAsync(AGG, 0, (size_t)N * 128 * sizeof(float), stream);
    edge_agg<4><<<eaBlocks, TB, 0, stream>>>(src, dst, EL, ER, EM, DN, FEAT, AGG, E, H1);
    finalize_elu<<<nfBlocks, TB, 0, stream>>>(AGG, b1, HBUF, N);

    // ===== layer 2 (1 head, D=128, residual) =====
    gemm128_wmma_f32<<<dim3(8, N / 16), 32, 0, stream>>>(HBUF, W2, FEAT);
    attn_coef<<<(N + TB - 1) / TB, TB, 0, stream>>>(FEAT, al2, ar2, EL, ER, N, 1, 128);
    hipMemsetAsync(EM, 0, (size_t)N * sizeof(unsigned), stream);
    hipMemsetAsync(DN, 0, (size_t)N * sizeof(float), stream);
    edge_max<<<ehBlocks2, TB, 0, stream>>>(src, dst, EL, ER, EM, E, 1);
    edge_den<<<ehBlocks2, TB, 0, stream>>>(src, dst, EL, ER, EM, DN, E, 1);
    hipMemsetAsync(AGG, 0, (size_t)N * 128 * sizeof(float), stream);
    edge_agg<7><<<eaBlocks, TB, 0, stream>>>(src, dst, EL, ER, EM, DN, FEAT, AGG, E, 1);
    finalize_res<<<nfBlocks, TB, 0, stream>>>(AGG, HBUF, b2, FEAT, N);  // FEAT := logits

    // ===== pooling + ArcFace =====
    hipMemsetAsync(GS, 0, (size_t)G * 128 * sizeof(float), stream);
    hipMemsetAsync(GC, 0, (size_t)G * sizeof(float), stream);
    pool_acc<<<nfBlocks, TB, 0, stream>>>(FEAT, gid, GS, GC, N);
    emb_norm<<<(G + 63) / 64, 64, 0, stream>>>(GS, GC, XN, G);
    wa_norm_t<<<1, 128, 0, stream>>>(Wa, WNT);
    gemm128_wmma_f32<<<dim3(8, G / 16), 32, 0, stream>>>(XN, WNT, COS1);
    arcface<<<(G * 128 + TB - 1) / TB, TB, 0, stream>>>(COS1, lab, out, G);
}